// IS_FTT_multi_layer_relu_76209899701089
// MI455X (gfx1250) — compile-verified
//
#include <hip/hip_runtime.h>
#include <math.h>

typedef float v2f __attribute__((ext_vector_type(2)));
typedef float v8f __attribute__((ext_vector_type(8)));

// ---------------------------------------------------------------------------
// Stage A: one TT layer core.  y[b,ip,ic,im,iq,id] = relu(sum_n x[b,ip,n,iq]*w[ic,n,im,id])
// Bandwidth-bound (<=10 FMA per 4B store) -> plain VALU, weight staged in LDS.
// ---------------------------------------------------------------------------
__global__ void tt_layer_core_kern(const float* __restrict__ x, const float* __restrict__ w,
                                   float* __restrict__ y, int total,
                                   int P, int n, int Q, int xsb,
                                   int c, int m, int d) {
  __shared__ float wl[900];
  const int wcount = c * n * m * d;
  for (int i = threadIdx.x; i < wcount; i += blockDim.x) wl[i] = w[i];
  __syncthreads();
  int t = blockIdx.x * blockDim.x + threadIdx.x;
  if (t >= total) return;
  int id = t % d; int r = t / d;
  int iq = r % Q; r /= Q;
  int im = r % m; r /= m;
  int ic = r % c; r /= c;
  int ip = r % P; int ib = r / P;
  const float* xp = x + (size_t)ib * xsb + (size_t)ip * n * Q + iq;
  const float* wp = wl + ic * n * m * d + im * d + id;
  float acc = 0.f;
  for (int in = 0; in < n; ++in) acc += xp[(size_t)in * Q] * wp[in * m * d];
  y[t] = fmaxf(acc, 0.f);
}

// ---------------------------------------------------------------------------
// Stage B GEMM: G[b,m,ncol] = sum_{k<729} A(m,k) * B(k,ncol)
//   A(m,k) = A[b*As + (m/3)*2187 + 3k + m%3]   (m < M; padded rows clamped,
//            their garbage lands only in C rows >= M which are never stored)
//   B(k,n) = Bm[b*Bs + k*N + n]                (padded cols clamped likewise)
// One wave per 16x16 tile; V_WMMA_F32_16X16X4_F32 over K = 182 clean steps + 1
// masked tail step (k=728). No branches in the hot loop.
// ---------------------------------------------------------------------------
__global__ void wmma_km_kern(const float* __restrict__ A, const float* __restrict__ Bm,
                             float* __restrict__ G,
                             int M, int N, int As, int Bs, int Gs) {
  const int lane  = threadIdx.x;            // 0..31
  const int ntile = blockIdx.x;
  const int mtile = blockIdx.y;
  const int b     = blockIdx.z;
  const int col   = ntile * 16 + (lane & 15);
  const int mrow  = mtile * 16 + (lane & 15);
  const int khalf = (lane >> 4) << 1;       // 0 for lanes 0-15, 2 for lanes 16-31
  const int colc  = col  < N ? col  : N - 1;   // clamp: padded lanes read valid memory
  const int mrowc = mrow < M ? mrow : M - 1;
  const float* Ab = A  + (size_t)b * As + (mrowc / 3) * 2187 + (mrowc % 3);
  const float* Bb = Bm + (size_t)b * Bs + colc;
  const float* Ap = Ab + 3 * khalf;
  const float* Bp = Bb + (size_t)khalf * N;
  v8f cacc = {};
#pragma unroll 4
  for (int kk = 0; kk < 182; ++kk) {        // k = 0 .. 727, fully valid
    v2f a, bf;
    a.x  = Ap[0];
    a.y  = Ap[3];
    bf.x = Bp[0];
    bf.y = Bp[N];
    cacc = __builtin_amdgcn_wmma_f32_16x16x4_f32(
        false, a, false, bf, (short)0, cacc, false, false);
    Ap += 12;
    Bp += (size_t)4 * N;
  }
  {                                          // tail: only k = 728 is real
    const float am = (khalf == 0) ? 1.f : 0.f;   // uniform per half-wave, no divergence
    v2f a  = {Ab[3 * 728] * am, 0.f};
    v2f bf = {Bb[(size_t)728 * N] * am, 0.f};
    cacc = __builtin_amdgcn_wmma_f32_16x16x4_f32(
        false, a, false, bf, (short)0, cacc, false, false);
  }
  const int rbase = mtile * 16 + ((lane >> 4) << 3);
#pragma unroll
  for (int v = 0; v < 8; ++v) {
    const int row = rbase + v;
    if (row < M && col < N) G[(size_t)b * Gs + (size_t)row * N + col] = cacc[v];
  }
}

// ---------------------------------------------------------------------------
// Stage B small contraction: out[b, mh*MO+mo, q, dn] =
//   sum_{di<3, n<NN} G[b, mh*3+di, n*729+q] * W[((di*NN+n)*MO+mo)*3+dn]
// ---------------------------------------------------------------------------
__global__ void tt_apply_w_kern(const float* __restrict__ G, const float* __restrict__ W,
                                float* __restrict__ out, int total, int MH, int MO, int NN) {
  int t = blockIdx.x * blockDim.x + threadIdx.x;
  if (t >= total) return;
  int dn = t % 3; int r = t / 3;
  int q  = r % 729; r /= 729;
  int mm = r % (MH * MO); int b = r / (MH * MO);
  int mh = mm / MO, mo = mm % MO;
  const int ld = NN * 729;
  const float* Gb = G + (size_t)b * (MH * 3) * ld + (size_t)(mh * 3) * ld + q;
  float acc = 0.f;
  for (int di = 0; di < 3; ++di)
    for (int n = 0; n < NN; ++n)
      acc += Gb[(size_t)di * ld + n * 729] * W[((di * NN + n) * MO + mo) * 3 + dn];
  out[t] = acc;
}

// u1[b,q1,d1] = sum_{n<4} X51[b,0,n,q1] * W61[0,n,0,d1]
__global__ void u1_kern(const float* __restrict__ X, const float* __restrict__ W,
                        float* __restrict__ u1) {
  int t = blockIdx.x * blockDim.x + threadIdx.x;
  if (t >= 4374) return;
  int d1 = t % 3; int r = t / 3; int q1 = r % 729; int b = r / 729;
  float acc = 0.f;
  for (int n = 0; n < 4; ++n) acc += X[(size_t)(b * 4 + n) * 729 + q1] * W[n * 3 + d1];
  u1[t] = acc;
}

// v5[b, q4*3+d4] = sum_{n<4} X55[b,q4,n,0] * W65[d4,n,0,0]
__global__ void v5_kern(const float* __restrict__ X, const float* __restrict__ W,
                        float* __restrict__ v5) {
  int t = blockIdx.x * blockDim.x + threadIdx.x;
  if (t >= 4374) return;
  int i = t % 2187; int b = t / 2187;
  int q4 = i / 3, d4 = i % 3;
  float acc = 0.f;
  for (int n = 0; n < 4; ++n) acc += X[(size_t)(b * 729 + q4) * 4 + n] * W[d4 * 4 + n];
  v5[t] = acc;
}

// logits[b,m] = sum_{i<2187} r4[b,m,i] * v5[b,i]
__global__ void logits_kern(const float* __restrict__ r4, const float* __restrict__ v5,
                            float* __restrict__ lg) {
  __shared__ float s[256];
  const int b = blockIdx.x / 10, m = blockIdx.x % 10;
  const float* rp = r4 + (size_t)(b * 10 + m) * 2187;
  const float* vp = v5 + (size_t)b * 2187;
  float a = 0.f;
  for (int i = threadIdx.x; i < 2187; i += 256) a += rp[i] * vp[i];
  s[threadIdx.x] = a;
  __syncthreads();
  for (int o = 128; o > 0; o >>= 1) {
    if ((int)threadIdx.x < o) s[threadIdx.x] += s[threadIdx.x + o];
    __syncthreads();
  }
  if (threadIdx.x == 0) lg[b * 10 + m] = s[0];
}

__global__ void logsm_kern(const float* __restrict__ lg, float* __restrict__ out) {
  int b = threadIdx.x;
  if (b < 2) {
    float mx = lg[b * 10];
    for (int j = 1; j < 10; ++j) mx = fmaxf(mx, lg[b * 10 + j]);
    float s = 0.f;
    for (int j = 0; j < 10; ++j) s += __expf(lg[b * 10 + j] - mx);
    float l = __logf(s);
    for (int j = 0; j < 10; ++j) out[b * 10 + j] = lg[b * 10 + j] - mx - l;
  }
}

// ---------------------------------------------------------------------------
extern "C" void kernel_launch(void* const* d_in, const int* in_sizes, int n_in,
                              void* d_out, int out_size, void* d_ws, size_t ws_size,
                              hipStream_t stream) {
  (void)in_sizes; (void)n_in; (void)out_size; (void)ws_size;
  const float* x = (const float*)d_in[0];
  const float* W[6][5];
  for (int l = 0; l < 6; ++l)
    for (int k = 0; k < 5; ++k) W[l][k] = (const float*)d_in[1 + l * 5 + k];
  float* ws = (float*)d_ws;

  const int IN[5]  = {3, 4, 8, 4, 8};
  const int MID[5] = {4, 5, 10, 5, 4};
  const int carr[5] = {1, 3, 3, 3, 3};
  const int darr[5] = {3, 3, 3, 3, 1};
  const int xoffs[5] = {0, 9, 45, 117, 153};

  const float* xin[5];
  int P[5] = {1, 3, 3, 3, 3};
  int Q[5] = {3, 3, 3, 3, 1};
  int NN[5], XSB[5];
  for (int k = 0; k < 5; ++k) { xin[k] = x + xoffs[k]; NN[k] = IN[k]; XSB[k] = 177; }

  size_t off = 0;
  // -------- Stage A: layers 1..5 with ReLU --------
  for (int l = 0; l < 5; ++l) {
    float* xout[5];
    for (int k = 0; k < 5; ++k) {
      const int c = carr[k], d = darr[k], m = MID[k];
      const int total = 2 * P[k] * c * m * Q[k] * d;
      xout[k] = ws + off; off += (size_t)total;
      const int blocks = (total + 255) / 256;
      hipLaunchKernelGGL(tt_layer_core_kern, dim3(blocks), dim3(256), 0, stream,
                         xin[k], W[l][k], xout[k], total,
                         P[k], NN[k], Q[k], XSB[k], c, m, d);
    }
    for (int k = 0; k < 5; ++k) {
      xin[k] = xout[k];
      P[k] *= carr[k];
      Q[k] *= darr[k];
      NN[k] = MID[k];
      XSB[k] = P[k] * NN[k] * Q[k];
    }
  }
  // xin[k] now = layer-5 outputs X5k: [2,1,4,729],[2,729,5,729],[2,729,10,729],[2,729,5,729],[2,729,4,1]

  // -------- Stage B workspace --------
  float* u1 = ws + off; off += 4374;
  float* G2 = ws + off; off += (size_t)2 * 3 * 3645;
  float* r2 = ws + off; off += 8748;
  float* G3 = ws + off; off += (size_t)2 * 6 * 7290;
  float* r3 = ws + off; off += 43740;
  float* G4 = ws + off; off += (size_t)2 * 30 * 3645;
  float* r4 = ws + off; off += 43740;
  float* v5 = ws + off; off += 4374;
  float* lg = ws + off; off += 32;

  // u1 = X51 contracted with W6_1
  hipLaunchKernelGGL(u1_kern, dim3((4374 + 255) / 256), dim3(256), 0, stream,
                     xin[0], W[5][0], u1);

  // G2 = u1^T * X52 : M=3, N=3645, K=729
  hipLaunchKernelGGL(wmma_km_kern, dim3(228, 1, 2), dim3(32), 0, stream,
                     u1, xin[1], G2, 3, 3645, 2187, 2657205, 3 * 3645);
  // r2: MH=1, MO=2, NN=5 -> 8748 outputs
  hipLaunchKernelGGL(tt_apply_w_kern, dim3((8748 + 255) / 256), dim3(256), 0, stream,
                     G2, W[5][1], r2, 8748, 1, 2, 5);

  // G3 = r2 * X53 : M=6, N=7290, K=729
  hipLaunchKernelGGL(wmma_km_kern, dim3(456, 1, 2), dim3(32), 0, stream,
                     r2, xin[2], G3, 6, 7290, 4374, 5314410, 6 * 7290);
  // r3: MH=2, MO=5, NN=10 -> 43740 outputs
  hipLaunchKernelGGL(tt_apply_w_kern, dim3((43740 + 255) / 256), dim3(256), 0, stream,
                     G3, W[5][2], r3, 43740, 2, 5, 10);

  // G4 = r3 * X54 : M=30, N=3645, K=729
  hipLaunchKernelGGL(wmma_km_kern, dim3(228, 2, 2), dim3(32), 0, stream,
                     r3, xin[3], G4, 30, 3645, 21870, 2657205, 30 * 3645);
  // r4: MH=10, MO=1, NN=5 -> 43740 outputs
  hipLaunchKernelGGL(tt_apply_w_kern, dim3((43740 + 255) / 256), dim3(256), 0, stream,
                     G4, W[5][3], r4, 43740, 10, 1, 5);

  // v5 = X55 contracted with W6_5
  hipLaunchKernelGGL(v5_kern, dim3((4374 + 255) / 256), dim3(256), 0, stream,
                     xin[4], W[5][4], v5);

  // logits + log_softmax
  hipLaunchKernelGGL(logits_kern, dim3(20), dim3(256), 0, stream, r4, v5, lg);
  hipLaunchKernelGGL(logsm_kern, dim3(1), dim3(32), 0, stream, lg, (float*)d_out);
}